// LMKAN_2D_Layer_66640712565397
// MI455X (gfx1250) — compile-verified
//
#include <hip/hip_runtime.h>
#include <stdint.h>

typedef float v2f  __attribute__((ext_vector_type(2)));
typedef float v8f  __attribute__((ext_vector_type(8)));
typedef unsigned int u32x4 __attribute__((ext_vector_type(4)));
typedef int   i32x8 __attribute__((ext_vector_type(8)));
typedef int   i32x4 __attribute__((ext_vector_type(4)));

#define NCH 64          // N_CHUNKS
#define DD  128         // D
#define OO  256         // O
#define BB  4096        // B
#define PP  64          // P = D/2
#define BT  16          // b-tile per block
#define GG  65          // grid points = NCH+1
#define CELL_STRIDE (OO * PP)   // floats per (i,j) cell of func_parameter

// Giles (2010) single-precision erfinv approximation.
__device__ __forceinline__ float erfinv_poly(float x) {
    float w = -__logf((1.0f - x) * (1.0f + x));
    float p;
    if (w < 5.0f) {
        w = w - 2.5f;
        p = 2.81022636e-08f;
        p = fmaf(p, w, 3.43273939e-07f);
        p = fmaf(p, w, -3.5233877e-06f);
        p = fmaf(p, w, -4.39150654e-06f);
        p = fmaf(p, w, 0.00021858087f);
        p = fmaf(p, w, -0.00125372503f);
        p = fmaf(p, w, -0.00417768164f);
        p = fmaf(p, w, 0.246640727f);
        p = fmaf(p, w, 1.50140941f);
    } else {
        w = __fsqrt_rn(w) - 3.0f;
        p = -0.000200214257f;
        p = fmaf(p, w, 0.000100950558f);
        p = fmaf(p, w, 0.00134934322f);
        p = fmaf(p, w, -0.00367342844f);
        p = fmaf(p, w, 0.00573950773f);
        p = fmaf(p, w, -0.0076224613f);
        p = fmaf(p, w, 0.00943887047f);
        p = fmaf(p, w, 1.00167406f);
        p = fmaf(p, w, 2.83297682f);
    }
    return p * x;
}

// Largest i in [0,63] with borders[i] <= x (x pre-clipped into [borders[0], borders[64]) ).
__device__ __forceinline__ int bucket_search(const float* b, float x) {
    int lo = 0, hi = NCH;
    #pragma unroll
    for (int s = 0; s < 6; ++s) {   // 2^6 = 64 intervals
        int mid = (lo + hi) >> 1;
        if (b[mid] <= x) lo = mid; else hi = mid;
    }
    return lo;
}

__global__ __launch_bounds__(256, 4)   // cap VGPRs <= 256: no vgpr-msb swapping, 4 waves/SIMD
void lmkan_fused_kernel(const float* __restrict__ x,
                        const float* __restrict__ wK,
                        const int*   __restrict__ relu_flag,
                        const float* __restrict__ fp,
                        const float* __restrict__ scale,
                        const float* __restrict__ biasd,
                        const float* __restrict__ W,
                        const float* __restrict__ biasL,
                        float* __restrict__ out)
{
    __shared__ float  xs[DD * BT];       // raw x tile, [d][bl], 8 KB
    __shared__ float  borders[GG];
    __shared__ int    cellp[PP * BT];    // packed (i<<8)|j per (p, bl)
    __shared__ float4 wts[PP * BT];      // bilinear corner weights per (p, bl)

    const int tid = threadIdx.x;
    const int b0  = blockIdx.x * BT;

    // ---- borders (CDF grid) ----
    if (tid < GG) {
        float pq = (float)tid / (float)NCH;
        pq = fminf(fmaxf(pq, 0.5f / NCH), 1.0f - 0.5f / NCH);
        borders[tid] = 1.41421356237f * erfinv_poly(2.0f * pq - 1.0f);
    }

    // Warm the L2 path for this wave's slice of W (global_prefetch_b8).
    __builtin_prefetch(W + (size_t)tid * DD, 0, 0);

    // ---- TDM async load of the 128x16 f32 x-tile into LDS (wave 0 issues) ----
    if ((tid >> 5) == 0) {
        uint64_t ga      = (uint64_t)(uintptr_t)(x + b0);
        uint32_t lds_off = (uint32_t)(uintptr_t)(&xs[0]);   // low 32 bits = LDS byte offset

        u32x4 g0;
        g0.x = 1u;                                             // count=1, user descriptor
        g0.y = lds_off;                                        // lds_addr
        g0.z = (uint32_t)ga;                                   // global_addr[31:0]
        g0.w = (uint32_t)((ga >> 32) & 0x1FFFFFFu) | (2u << 30); // global_addr[56:32] | type=2

        i32x8 g1;
        g1[0] = (2 << 16);                                     // data_size = 4 B
        g1[1] = (int)((BB & 0xFFFF) << 16);                    // tensor_dim0[15:0] (=B along b)
        g1[2] = (int)(((BB >> 16) & 0xFFFF) | ((DD & 0xFFFF) << 16)); // dim0 hi | dim1 lo (=D)
        g1[3] = (int)(((DD >> 16) & 0xFFFF) | (BT << 16));     // dim1 hi | tile_dim0 = 16
        g1[4] = (int)(DD & 0xFFFF);                            // tile_dim1 = 128, tile_dim2 = 0
        g1[5] = (int)BB;                                       // tensor_dim0_stride low (=B)
        g1[6] = 0;                                             // stride hi | dim1_stride lo
        g1[7] = 0;

        i32x4 z4; z4[0] = 0; z4[1] = 0; z4[2] = 0; z4[3] = 0;
        i32x8 z8; z8[0] = 0; z8[1] = 0; z8[2] = 0; z8[3] = 0;
                  z8[4] = 0; z8[5] = 0; z8[6] = 0; z8[7] = 0;
        __builtin_amdgcn_tensor_load_to_lds(g0, g1, z4, z4, z8, 0);
        __builtin_amdgcn_s_wait_tensorcnt(0);
    }
    __syncthreads();

    // ---- per-(p, bl) bucket indices + bilinear weights (computed once, reused by 8 waves) ----
    const float blo = borders[0];
    const float bhi = borders[NCH] - 1e-6f;
    for (int e = tid; e < PP * BT; e += 256) {
        int p  = e >> 4;
        int bl = e & 15;
        float x1 = fmaf(xs[(2 * p)     * BT + bl], scale[2 * p],     biasd[2 * p]);
        float x2 = fmaf(xs[(2 * p + 1) * BT + bl], scale[2 * p + 1], biasd[2 * p + 1]);
        x1 = fminf(fmaxf(x1, blo), bhi);
        x2 = fminf(fmaxf(x2, blo), bhi);
        int i = bucket_search(borders, x1);
        int j = bucket_search(borders, x2);
        float t1 = (x1 - borders[i]) / (borders[i + 1] - borders[i]);
        float t2 = (x2 - borders[j]) / (borders[j + 1] - borders[j]);
        cellp[e] = (i << 8) | j;
        wts[e] = make_float4((1.0f - t1) * (1.0f - t2), t1 * (1.0f - t2),
                             (1.0f - t1) * t2,          t1 * t2);
    }
    __syncthreads();

    // ---- per-wave: two 16x16 output tiles (o rows [obase, obase+32)) x 16 b columns ----
    const int wv   = tid >> 5;       // wave 0..7
    const int lane = tid & 31;
    const int n    = lane & 15;      // output column within tile (also A-row index)
    const int kh   = lane >> 4;      // lane half -> K pairing / C-row offset
    const int obase = wv * 32;

    // Linear part: exact f32 GEMM via V_WMMA_F32_16X16X4_F32, K-steps of 4 over D=128.
    // A(16x4): M = lane&15, component c holds K = 2*kh + c.
    // B(4x16): N = lane&15, component c holds K = 2*kh + c.
    // Partial unroll: keeps the (non-critical) GEMM pipelined without holding
    // all 96 A/B float2 operands live at once (register pressure).
    v8f c0 = {};
    v8f c1 = {};
    #pragma unroll 8
    for (int k = 0; k < DD; k += 4) {
        const int kk = k + 2 * kh;
        v2f bv; bv.x = xs[kk * BT + n]; bv.y = xs[(kk + 1) * BT + n];
        const float* wrow0 = W + (size_t)(obase + n) * DD + kk;
        v2f a0; a0.x = wrow0[0]; a0.y = wrow0[1];
        const float* wrow1 = wrow0 + (size_t)16 * DD;
        v2f a1; a1.x = wrow1[0]; a1.y = wrow1[1];
        c0 = __builtin_amdgcn_wmma_f32_16x16x4_f32(false, a0, false, bv, (short)0, c0, false, false);
        c1 = __builtin_amdgcn_wmma_f32_16x16x4_f32(false, a1, false, bv, (short)0, c1, false, false);
    }

    // lmKAN gather-accumulate: lane owns column n, o rows { obase + t*16 + v + 8*kh }.
    float acc[16];
    #pragma unroll
    for (int q = 0; q < 16; ++q) acc[q] = 0.0f;

    for (int p = 0; p < PP; ++p) {
        const int    ij = cellp[p * BT + n];
        const float4 w4 = wts[p * BT + n];
        const int ci = ij >> 8;
        const int cj = ij & 0xFF;
        const float* base = fp + (size_t)(ci * GG + cj) * CELL_STRIDE + p;
        #pragma unroll
        for (int t = 0; t < 2; ++t) {
            #pragma unroll
            for (int v = 0; v < 8; ++v) {
                const int o = obase + t * 16 + v + 8 * kh;
                const size_t off = (size_t)o * PP;
                float g00 = base[off];
                float g10 = base[off + (size_t)GG * CELL_STRIDE];        // (i+1, j)
                float g01 = base[off + (size_t)CELL_STRIDE];             // (i, j+1)
                float g11 = base[off + (size_t)(GG + 1) * CELL_STRIDE];  // (i+1, j+1)
                acc[t * 8 + v] += w4.x * g00 + w4.y * g10 + w4.z * g01 + w4.w * g11;
            }
        }
    }

    // ---- combine: relu(Wx+b) + weight_lmKAN * lmkan, store (O,B) row-major ----
    const float wKs  = wK[0];
    const int   relu = relu_flag[0];
    #pragma unroll
    for (int t = 0; t < 2; ++t) {
        #pragma unroll
        for (int v = 0; v < 8; ++v) {
            const int o = obase + t * 16 + v + 8 * kh;
            float lin = (t ? c1[v] : c0[v]) + biasL[o];
            if (relu) lin = fmaxf(lin, 0.0f);
            out[(size_t)o * BB + b0 + n] = lin + wKs * acc[t * 8 + v];
        }
    }
}

extern "C" void kernel_launch(void* const* d_in, const int* in_sizes, int n_in,
                              void* d_out, int out_size, void* d_ws, size_t ws_size,
                              hipStream_t stream) {
    (void)in_sizes; (void)n_in; (void)out_size; (void)d_ws; (void)ws_size;
    const float* x      = (const float*)d_in[0];   // (D, B)
    const float* wK     = (const float*)d_in[1];   // scalar
    const int*   relu   = (const int*)  d_in[2];   // scalar flag
    const float* fp     = (const float*)d_in[3];   // (65, 65, O, P)
    const float* scale  = (const float*)d_in[4];   // (D,)
    const float* biasd  = (const float*)d_in[5];   // (D,)
    const float* W      = (const float*)d_in[6];   // (O, D)
    const float* biasL  = (const float*)d_in[7];   // (O,)
    float*       out    = (float*)d_out;           // (O, B)

    dim3 grid(BB / BT);   // 256 blocks
    dim3 block(256);      // 8 waves
    lmkan_fused_kernel<<<grid, block, 0, stream>>>(x, wK, relu, fp, scale, biasd, W, biasL, out);
}